// FilterDetections_53025666237041
// MI455X (gfx1250) — compile-verified
//
#include <hip/hip_runtime.h>
#include <stdint.h>

#define B_       8
#define N_       50000
#define C_       90
#define MAX_DET  300
#define NEGV     (-1e30f)
#define OK_THR   (-1e29f)
#define SCORE_THR 0.05f
#define IOU_THR   0.5f
#define EPSV      1e-8f

typedef unsigned int u32x4 __attribute__((ext_vector_type(4)));
typedef int          i32x8 __attribute__((ext_vector_type(8)));
typedef int          i32x4 __attribute__((ext_vector_type(4)));

// ---- wave32 argmax reduction with first-occurrence tie-break --------------
__device__ __forceinline__ void wave_argmax(float& v, int& i) {
#pragma unroll
  for (int off = 16; off > 0; off >>= 1) {
    float ov = __shfl_down(v, off, 32);
    int   oi = __shfl_down(i, off, 32);
    if (ov > v || (ov == v && oi < i)) { v = ov; i = oi; }
  }
}

#if !__has_builtin(__builtin_amdgcn_tensor_load_to_lds)
#warning "CDNA5 probe: __builtin_amdgcn_tensor_load_to_lds NOT available -> plain LDS copy fallback in use"
#endif
#if !__has_builtin(__builtin_amdgcn_s_wait_tensorcnt)
#warning "CDNA5 probe: __builtin_amdgcn_s_wait_tensorcnt NOT available"
#endif

// ---- One TDM descriptor: contiguous 1-D copy of n f32 from global -> LDS --
#if __has_builtin(__builtin_amdgcn_tensor_load_to_lds)
__device__ __forceinline__ void tdm_load_f32(float* lds_dst, const float* gsrc,
                                             unsigned n) {
  unsigned long long ga = (unsigned long long)(uintptr_t)gsrc;
  unsigned lds_off = (unsigned)(uintptr_t)lds_dst;   // low 32 bits = LDS offset
  u32x4 g0;
  g0.x = 1u;                                         // count=1 (valid), user mode
  g0.y = lds_off;                                    // lds_addr (bytes)
  g0.z = (unsigned)(ga & 0xFFFFFFFFu);               // global_addr[31:0]
  g0.w = (unsigned)((ga >> 32) & 0x1FFFFFFu) | (2u << 30);  // addr[56:32]|type=2
  i32x8 g1;
  g1[0] = (int)(2u << 16);                           // data_size=4B, no flags
  g1[1] = (int)((n & 0xFFFFu) << 16);                // tensor_dim0[15:0]
  g1[2] = (int)((n >> 16) | (1u << 16));             // dim0[31:16] | tensor_dim1=1
  g1[3] = (int)((n & 0xFFFFu) << 16);                // dim1[31:16]=0 | tile_dim0=n
  g1[4] = 1;                                         // tile_dim1=1, tile_dim2=0
  g1[5] = (int)n;                                    // tensor_dim0_stride[31:0]
  g1[6] = (int)((n & 0xFFFFu) << 16);                // stride0 hi=0 | stride1[15:0]
  g1[7] = 0;                                         // stride1[47:16]=0
  i32x4 g2 = {0, 0, 0, 0};
  i32x4 g3 = {0, 0, 0, 0};
#if defined(__clang_major__) && (__clang_major__ >= 23)
  i32x8 g4 = {0, 0, 0, 0, 0, 0, 0, 0};
  __builtin_amdgcn_tensor_load_to_lds(g0, g1, g2, g3, g4, 0);
#else
  __builtin_amdgcn_tensor_load_to_lds(g0, g1, g2, g3, 0);
#endif
}
#endif

// ===========================================================================
// Phase 1: per-row max/argmax over C=90 classes. One wave per row.
// 144 MB streamed once from HBM -> non-temporal loads (keep L2 for boxes).
// ===========================================================================
__global__ void class_reduce_kernel(const float* __restrict__ cls,
                                    float* __restrict__ s0,
                                    int*   __restrict__ lab) {
  const int wid  = threadIdx.x >> 5;
  const int lane = threadIdx.x & 31;
  const long long row = (long long)blockIdx.x * 8 + wid;   // exactly B*N rows
  const float* p = cls + row * C_;

  float best = -3.402823466e38f;
  int   bidx = 0;
#pragma unroll
  for (int k = 0; k < 3; ++k) {
    int c = lane + 32 * k;
    if (c < C_) {
      float v = __builtin_nontemporal_load(p + c);
      if (v > best) { best = v; bidx = c; }   // ascending c -> first occurrence
    }
  }
  wave_argmax(best, bidx);
  if (lane == 0) {
    lab[row] = bidx;
    s0[row]  = (best > SCORE_THR) ? best : NEGV;
  }
}

// ===========================================================================
// Phase 2: greedy NMS, one workgroup (1024 thr = 32 waves) per batch.
// Scores (195 KB) live in LDS, staged by 4 Tensor Data Mover descriptors.
// Boxes (800 KB/batch) stay L2-resident; dead-lane skip + prefetch in loop.
// ===========================================================================
__global__ void __launch_bounds__(1024)
nms_kernel(const float* __restrict__ boxes,     // [B,N,4]
           const float* __restrict__ s0,        // [B,N] thresholded scores
           const int*   __restrict__ lab,       // [B,N]
           float*       __restrict__ out) {
  extern __shared__ __align__(16) char smem_raw[];
  float* s_sc = (float*)smem_raw;               // N_ floats (200000 B)

  __shared__ float r_val[32];
  __shared__ int   r_idx[32];
  __shared__ int   sel_idx[MAX_DET];
  __shared__ float sel_sc[MAX_DET];
  __shared__ float s_bb[4];

  const int tid  = threadIdx.x;
  const int lane = tid & 31;
  const int wid  = tid >> 5;
  const int b    = blockIdx.x;
  const float4* __restrict__ bx4 = (const float4*)boxes + (size_t)b * N_;
  const float*  __restrict__ src = s0 + (size_t)b * N_;

  // ---- Stage scores into LDS: 4 pipelined TDM descriptors from wave 0 ----
#if __has_builtin(__builtin_amdgcn_tensor_load_to_lds)
  if (wid == 0) {
    const unsigned Q = N_ / 4;                  // 12500 f32 per descriptor
    tdm_load_f32(s_sc + 0 * Q, src + 0 * Q, Q);
    tdm_load_f32(s_sc + 1 * Q, src + 1 * Q, Q);
    tdm_load_f32(s_sc + 2 * Q, src + 2 * Q, Q);
    tdm_load_f32(s_sc + 3 * Q, src + 3 * Q, Q);
    __builtin_amdgcn_s_wait_tensorcnt(0);
  }
#else
  for (int i = tid; i < N_; i += 1024) s_sc[i] = src[i];
#endif
  __syncthreads();

  // ---- 300 sequential greedy-NMS steps ----
  for (int t = 0; t < MAX_DET; ++t) {
    // 1) strided local argmax over LDS scores (ascending i -> first occurrence)
    float best = -3.402823466e38f;
    int   bidx = 0;
    for (int i = tid; i < N_; i += 1024) {
      float v = s_sc[i];
      if (v > best) { best = v; bidx = i; }
    }
    wave_argmax(best, bidx);
    if (lane == 0) { r_val[wid] = best; r_idx[wid] = bidx; }
    __syncthreads();

    // 2) cross-wave reduce in wave 0; broadcast selected box through LDS
    if (tid < 32) {
      best = r_val[tid];
      bidx = r_idx[tid];
      wave_argmax(best, bidx);
      if (tid == 0) {
        sel_idx[t] = bidx;
        sel_sc[t]  = best;
        float4 q = bx4[bidx];
        s_bb[0] = q.x; s_bb[1] = q.y; s_bb[2] = q.z; s_bb[3] = q.w;
        s_sc[bidx] = NEGV;                    // .at[i].set(NEG)
      }
    }
    __syncthreads();

    // 3) suppress IoU > 0.5 against the selected box
    const float bx1 = s_bb[0], by1 = s_bb[1], bx2 = s_bb[2], by2 = s_bb[3];
    const float barea = fmaxf(bx2 - bx1, 0.0f) * fmaxf(by2 - by1, 0.0f);
    for (int i = tid; i < N_; i += 1024) {
      __builtin_prefetch(bx4 + i + 1024, 0, 0);   // speculative; OOB is dropped
      float sv = s_sc[i];
      if (sv == NEGV) continue;               // already dead: NEG stays NEG
      float4 q = bx4[i];
      float iw = fmaxf(fminf(bx2, q.z) - fmaxf(bx1, q.x), 0.0f);
      float ih = fmaxf(fminf(by2, q.w) - fmaxf(by1, q.y), 0.0f);
      float inter = iw * ih;
      float area = fmaxf(q.z - q.x, 0.0f) * fmaxf(q.w - q.y, 0.0f);
      float iou = inter / (barea + area - inter + EPSV);
      if (iou > IOU_THR) s_sc[i] = NEGV;
    }
    __syncthreads();
  }

  // ---- gather outputs: boxes f32 | scores f32 | labels i32 (bit layout) ----
  const int* labp = lab + (size_t)b * N_;
  float*  ob = out;                              // [B,300,4] f32
  float*  os = out + (size_t)B_ * MAX_DET * 4;   // [B,300]   f32
  int*    ol = (int*)(out + (size_t)B_ * MAX_DET * 5);  // [B,300] i32
  for (int t = tid; t < MAX_DET; t += 1024) {
    int   idx = sel_idx[t];
    float sc  = sel_sc[t];
    bool  ok  = sc > OK_THR;
    float4 q  = bx4[idx];
    float4 r;
    r.x = ok ? q.x : -1.0f; r.y = ok ? q.y : -1.0f;
    r.z = ok ? q.z : -1.0f; r.w = ok ? q.w : -1.0f;
    ((float4*)ob)[(size_t)b * MAX_DET + t] = r;
    os[(size_t)b * MAX_DET + t] = ok ? sc : -1.0f;
    ol[(size_t)b * MAX_DET + t] = ok ? labp[idx] : -1;
  }
}

// ===========================================================================
extern "C" void kernel_launch(void* const* d_in, const int* in_sizes, int n_in,
                              void* d_out, int out_size, void* d_ws, size_t ws_size,
                              hipStream_t stream) {
  const float* boxes = (const float*)d_in[0];          // [B,N,4] f32
  const float* cls   = (const float*)d_in[1];          // [B,N,C] f32
  float* out = (float*)d_out;

  float* ws_s0  = (float*)d_ws;                                    // B*N f32
  int*   ws_lab = (int*)((char*)d_ws + (size_t)B_ * N_ * sizeof(float));

  // Phase 1: 8 waves/block, one wave per row -> B*N/8 blocks.
  class_reduce_kernel<<<(B_ * N_) / 8, 256, 0, stream>>>(cls, ws_s0, ws_lab);

  // Phase 2: one 1024-thread workgroup per batch; 200 KB dynamic LDS.
  size_t shmem = (size_t)N_ * sizeof(float);
  (void)hipFuncSetAttribute((const void*)nms_kernel,
                            hipFuncAttributeMaxDynamicSharedMemorySize,
                            (int)shmem);
  nms_kernel<<<B_, 1024, shmem, stream>>>(boxes, ws_s0, ws_lab, out);
  (void)in_sizes; (void)n_in; (void)out_size; (void)ws_size;
}